// RTDetrHungarianMatcher_69561290326385
// MI455X (gfx1250) — compile-verified
//
// RT-DETR Hungarian matcher cost matrix for MI455X (gfx1250, wave32).
// out[i,j] = 5*L1(box_i,box_j) + 2*focal(sigmoid(logit[i,label_j])) - 2*GIoU(i,j)
// Strategy:
//  * per-query 80-class focal-cost table (kills 40x of the log/exp work)
//  * class gather done as exact one-hot f32 WMMA (V_WMMA_F32_16X16X4_F32, K=80 -> 20 ops/tile)
//  * target records staged to LDS with global_load_async_to_lds_b64, double buffered,
//    synced via s_wait_asynccnt + barriers; DMA overlaps transcendental table build
//  * query data register-resident; 123 MB output streamed (write-bandwidth roofline)
#include <hip/hip_runtime.h>
#include <hip/hip_bf16.h>
#include <cstdint>

#ifndef __has_builtin
#define __has_builtin(x) 0
#endif

namespace {
constexpr int kNC   = 80;
constexpr int kNT   = 3200;
constexpr int kNQT  = 32 * 300;          // 9600 flattened query rows
constexpr float kAlpha  = 0.25f;
constexpr float kWClass = 2.0f, kWBbox = 5.0f, kWGiou = 2.0f;

constexpr int QT     = 16;               // queries per block (WMMA M)
constexpr int CHUNK  = 128;              // targets per LDS chunk = 8 tiles of 16
constexpr int NCHUNK = kNT / CHUNK;      // 25
constexpr int TDS    = 12;               // dwords per staged target record (b128-friendly)
constexpr int CTS    = 84;               // class-table LDS stride (80+pad, conflict-free)
}

typedef float v2f __attribute__((ext_vector_type(2)));
typedef float v8f __attribute__((ext_vector_type(8)));
typedef int   v2i __attribute__((vector_size(8)));   // matches builtin's b64 payload type

__device__ __forceinline__ float fast_rcp(float x) { return __builtin_amdgcn_rcpf(x); }

// ---- CDNA5 async global->LDS copy (ASYNCcnt-tracked) --------------------
__device__ __forceinline__ void async_copy_b64(const float* g, float* l) {
#if __has_builtin(__builtin_amdgcn_global_load_async_to_lds_b64)
  typedef __attribute__((address_space(1))) v2i* gptr_t;   // prints as "__device__" in HIP
  typedef __attribute__((address_space(3))) v2i* lptr_t;   // LDS: 32-bit wave-relative offset
  __builtin_amdgcn_global_load_async_to_lds_b64(
      (gptr_t)(uintptr_t)g, (lptr_t)(uintptr_t)l, 0, 0);
#else
  asm volatile("global_load_async_to_lds_b64 %0, %1, off"
               :: "v"((uint32_t)(uintptr_t)l), "v"(g) : "memory");
#endif
}

template <int N>
__device__ __forceinline__ void wait_asynccnt() {
#if __has_builtin(__builtin_amdgcn_s_wait_asynccnt)
  __builtin_amdgcn_s_wait_asynccnt(N);
#else
  if constexpr (N == 0) asm volatile("s_wait_asynccnt 0x0" ::: "memory");
  else                  asm volatile("s_wait_asynccnt 0x3" ::: "memory");
#endif
}

// ---- phase A: build 12-dword target records in workspace ----------------
// [cx,cy,w,h, x0,y0,x1,y1, area, label_bits, 0, 0]
__global__ void rtdetr_tprep_kernel(const float* __restrict__ tboxes,
                                    const int*   __restrict__ tlabels,
                                    float*       __restrict__ tdata) {
  int t = blockIdx.x * blockDim.x + threadIdx.x;
  if (t >= kNT) return;
  float cx = tboxes[4 * t + 0], cy = tboxes[4 * t + 1];
  float w  = tboxes[4 * t + 2], h  = tboxes[4 * t + 3];
  float x0 = cx - 0.5f * w, y0 = cy - 0.5f * h;
  float x1 = cx + 0.5f * w, y1 = cy + 0.5f * h;
  float* r = tdata + t * TDS;
  r[0] = cx; r[1] = cy; r[2] = w;  r[3] = h;
  r[4] = x0; r[5] = y0; r[6] = x1; r[7] = y1;
  r[8] = (x1 - x0) * (y1 - y0);
  r[9] = __int_as_float(tlabels[t]);
  r[10] = 0.0f; r[11] = 0.0f;
}

// ---- phase B: 16 queries x 3200 targets per block -----------------------
__launch_bounds__(256)
__global__ void rtdetr_cost_kernel(const float* __restrict__ logits,
                                   const float* __restrict__ pboxes,
                                   const float* __restrict__ tdata,
                                   float*       __restrict__ out) {
  __shared__ __attribute__((aligned(16))) float sCtab[QT * CTS];      // 5.25 KB
  __shared__ __attribute__((aligned(16))) float sQd[QT * 9];
  __shared__ __attribute__((aligned(16))) float sTd[2][CHUNK * TDS];  // 12 KB

  const int tid   = threadIdx.x;
  const int lane  = tid & 31;
  const int wv    = tid >> 5;          // wave id 0..7 -> tile within chunk
  const int half  = lane >> 4;         // WMMA half-wave
  const int nn    = lane & 15;         // WMMA N (and A-row M)
  const int qbase = blockIdx.x * QT;

  // Kick off DMA of target chunk 0 immediately; it overlaps the exp/log below.
  {
    const float* src = tdata;          // chunk 0
#pragma unroll
    for (int k = 0; k < 3; ++k) {
      int j = tid + 256 * k;           // 0..767 b64 transfers (CHUNK*TDS/2)
      async_copy_b64(src + 2 * j, &sTd[0][2 * j]);
    }
  }

  // Per-query focal class-cost table: 16 x 80 (2 logs per entry).
#pragma unroll
  for (int it = 0; it < (QT * kNC) / 256; ++it) {
    int e = tid + 256 * it;
    int q = e / kNC, c = e - q * kNC;
    float x   = logits[(qbase + q) * kNC + c];
    float p   = fast_rcp(1.0f + __expf(-x));
    float omp = 1.0f - p;
    float pos = kAlpha          * omp * omp * (-__logf(p   + 1e-8f));
    float neg = (1.0f - kAlpha) * p   * p   * (-__logf(omp + 1e-8f));
    sCtab[q * CTS + c] = pos - neg;
  }
  // Per-query geometry.
  if (tid < QT) {
    int q = qbase + tid;
    float cx = pboxes[4 * q + 0], cy = pboxes[4 * q + 1];
    float w  = pboxes[4 * q + 2], h  = pboxes[4 * q + 3];
    float* r = &sQd[tid * 9];
    r[0] = cx; r[1] = cy; r[2] = w; r[3] = h;
    r[4] = cx - 0.5f * w; r[5] = cy - 0.5f * h;
    r[6] = cx + 0.5f * w; r[7] = cy + 0.5f * h;
    r[8] = w * h;
  }
  __syncthreads();

  // Query rows this lane owns (M = v + 8*half) -> registers for whole kernel.
  float qcx[8], qcy[8], qw[8], qh[8], qx0[8], qy0[8], qx1[8], qy1[8], qar[8];
#pragma unroll
  for (int v = 0; v < 8; ++v) {
    const float* r = &sQd[(v + 8 * half) * 9];
    qcx[v] = r[0]; qcy[v] = r[1]; qw[v] = r[2]; qh[v] = r[3];
    qx0[v] = r[4]; qy0[v] = r[5]; qx1[v] = r[6]; qy1[v] = r[7];
    qar[v] = r[8];
  }

  for (int c = 0; c < NCHUNK; ++c) {
    if (c + 1 < NCHUNK) {
      const float* src = tdata + (c + 1) * CHUNK * TDS;
      float* dst = sTd[(c + 1) & 1];
#pragma unroll
      for (int k = 0; k < 3; ++k) {
        int j = tid + 256 * k;
        async_copy_b64(src + 2 * j, &dst[2 * j]);
      }
      wait_asynccnt<3>();              // chunk c's 3 DMAs retired (in-order)
    } else {
      wait_asynccnt<0>();
    }
    __syncthreads();

    // This wave's 16-target tile within the chunk.
    const float* tr = &sTd[c & 1][(wv * 16 + nn) * TDS];
    const float4 t0 = *(const float4*)(tr);       // cx cy w h
    const float4 t1 = *(const float4*)(tr + 4);   // x0 y0 x1 y1
    const float  tar  = tr[8];
    const int    tlab = __float_as_int(tr[9]);

    // Exact class-cost gather as one-hot f32 WMMA: D = table(16x80) * onehot(80x16).
    v8f acc = {0.f, 0.f, 0.f, 0.f, 0.f, 0.f, 0.f, 0.f};
#pragma unroll
    for (int kk = 0; kk < kNC / 4; ++kk) {
      const int c0 = 4 * kk + 2 * half;           // K slots {2h, 2h+1} per ISA layout
      v2f a = *(const v2f*)&sCtab[nn * CTS + c0]; // A[M=nn, K]
      v2f b;
      b.x = (tlab == c0    ) ? 1.0f : 0.0f;       // B[K, N=nn]
      b.y = (tlab == c0 + 1) ? 1.0f : 0.0f;
      acc = __builtin_amdgcn_wmma_f32_16x16x4_f32(
          false, a, false, b, (short)0, acc, false, false);
    }

    // L1 + GIoU + combine; lane fixes target, sweeps its 8 query rows.
    const int col = c * CHUNK + wv * 16 + nn;
    float* orow = out + (qbase + 8 * half) * kNT + col;
#pragma unroll
    for (int v = 0; v < 8; ++v) {
      float l1 = fabsf(qcx[v] - t0.x) + fabsf(qcy[v] - t0.y) +
                 fabsf(qw[v]  - t0.z) + fabsf(qh[v]  - t0.w);
      float iw = fmaxf(fminf(qx1[v], t1.z) - fmaxf(qx0[v], t1.x), 0.0f);
      float ih = fmaxf(fminf(qy1[v], t1.w) - fmaxf(qy0[v], t1.y), 0.0f);
      float inter = iw * ih;
      float uni   = qar[v] + tar - inter;
      float ew = fmaxf(fmaxf(qx1[v], t1.z) - fminf(qx0[v], t1.x), 0.0f);
      float eh = fmaxf(fmaxf(qy1[v], t1.w) - fminf(qy0[v], t1.y), 0.0f);
      float ear = ew * eh;
      float giou = inter * fast_rcp(uni) - (ear - uni) * fast_rcp(ear);
      orow[v * kNT] = kWBbox * l1 + kWClass * acc[v] - kWGiou * giou;
    }
    __syncthreads();                   // WAR: buffer (c&1) is DMA target at c+1
  }
}

extern "C" void kernel_launch(void* const* d_in, const int* in_sizes, int n_in,
                              void* d_out, int out_size, void* d_ws, size_t ws_size,
                              hipStream_t stream) {
  const float* logits  = (const float*)d_in[0];   // (32,300,80) f32
  const float* pboxes  = (const float*)d_in[1];   // (32,300,4)  f32
  const int*   tlabels = (const int*)  d_in[2];   // (3200,)     i32
  const float* tboxes  = (const float*)d_in[3];   // (3200,4)    f32
  float* out   = (float*)d_out;                   // (32,300,3200) f32
  float* tdata = (float*)d_ws;                    // 3200*12*4 = 150 KB scratch

  rtdetr_tprep_kernel<<<(kNT + 255) / 256, 256, 0, stream>>>(tboxes, tlabels, tdata);
  rtdetr_cost_kernel<<<kNQT / QT, 256, 0, stream>>>(logits, pboxes, tdata, out);
}